// LSTMModel_58892591562893
// MI455X (gfx1250) — compile-verified
//
#include <hip/hip_runtime.h>
#include <math.h>

typedef __attribute__((ext_vector_type(2))) float v2f;
typedef __attribute__((ext_vector_type(8))) float v8f;

#define T_STEPS  262144
#define IN_SZ    86
#define HID      32
#define GATES    128   // 4*HID, torch gate order i|f|g|o

// ---------------------------------------------------------------------------
// Phase 1: x_gates[T,128] = input[T,86] @ W_ih^T[86,128] + (b_ih + b_hh)
// fp32 WMMA (V_WMMA_F32_16X16X4_F32), one wave per 16x16 output tile.
// Block = 256 threads = 8 waves -> one 16-row stripe x all 128 gate columns.
// Bandwidth-bound (~224 MB traffic); WMMA keeps full fp32 precision.
// ---------------------------------------------------------------------------
__global__ __launch_bounds__(256) void xgates_wmma_kernel(
    const float* __restrict__ input,   // [T, 86]
    const float* __restrict__ W_ih,    // [128, 86] row-major
    const float* __restrict__ b_ih,    // [128]
    const float* __restrict__ b_hh,    // [128]
    float* __restrict__ xg)            // [T, 128]
{
    const int lane = threadIdx.x & 31;
    const int wave = threadIdx.x >> 5;          // 0..7 : column tile
    const int row0 = blockIdx.x * 16;           // output row base
    const int col0 = wave * 16;                 // output col base

    const int half = lane >> 4;                 // 0: lanes 0-15, 1: lanes 16-31
    const int lr   = lane & 15;

    // A 16x4 f32 fragment: lanes 0-15 hold K={k,k+1}, lanes 16-31 K={k+2,k+3}
    // B 4x16 f32 fragment: mirrored (row of B striped across lanes per VGPR)
    const float* arow = input + (size_t)(row0 + lr) * IN_SZ;
    const float* wrow = W_ih  + (size_t)(col0 + lr) * IN_SZ;   // B[k][n] = W_ih[n][k]

    v8f acc = {};
    for (int k = 0; k < IN_SZ; k += 4) {
        const int ka = k + 2 * half;            // this lane-half's K offset
        v2f a, b;
        a.x = (ka + 0 < IN_SZ) ? arow[ka + 0] : 0.0f;
        a.y = (ka + 1 < IN_SZ) ? arow[ka + 1] : 0.0f;
        b.x = (ka + 0 < IN_SZ) ? wrow[ka + 0] : 0.0f;
        b.y = (ka + 1 < IN_SZ) ? wrow[ka + 1] : 0.0f;
        // D = A(16x4) * B(4x16) + C  -- emits v_wmma_f32_16x16x4_f32
        acc = __builtin_amdgcn_wmma_f32_16x16x4_f32(
                  false, a, false, b, (short)0, acc, false, false);
    }

    // C/D layout: VGPR v, lanes 0-15 -> M=v, N=lane; lanes 16-31 -> M=v+8
    const float bias = b_ih[col0 + lr] + b_hh[col0 + lr];
#pragma unroll
    for (int v = 0; v < 8; ++v) {
        const int m = v + 8 * half;
        xg[(size_t)(row0 + m) * GATES + col0 + lr] = acc[v] + bias;
    }
}

// ---------------------------------------------------------------------------
// Phase 2: sequential LSTM scan, ONE wave32 owns the whole recurrence.
// Lane l owns h[l], c[l] and gate rows {l, l+32, l+64, l+96} so i,f,g,o land
// on the owning lane with no cross-lane traffic. The 32x128 matvec broadcasts
// h[k] via v_readlane (SGPR) feeding v_fmac. W_hh lives in 128 VGPRs/lane.
// x_gates (134 MB) fits in the 192 MB L2, so per-step loads hit L2.
// ---------------------------------------------------------------------------
__device__ __forceinline__ float readlane_f(float v, int lane) {
    return __int_as_float(__builtin_amdgcn_readlane(__float_as_int(v), lane));
}
__device__ __forceinline__ float sigmoid_fast(float x) {
    return 1.0f / (1.0f + __expf(-x));
}

__global__ __launch_bounds__(32) void lstm_scan_kernel(
    const float* __restrict__ xg,      // [T, 128]
    const float* __restrict__ W_hh,    // [128, 32] row-major
    const float* __restrict__ fc_w,    // [1, 32]
    const float* __restrict__ fc_b,    // [1]
    float* __restrict__ out)           // [1]
{
    const int l = threadIdx.x;         // 0..31

    // Per-lane weight rows, kept in VGPRs.
    float wi[HID], wf[HID], wg[HID], wo[HID];
#pragma unroll
    for (int k = 0; k < HID; ++k) {
        wi[k] = W_hh[(size_t)(0 * HID + l) * HID + k];
        wf[k] = W_hh[(size_t)(1 * HID + l) * HID + k];
        wg[k] = W_hh[(size_t)(2 * HID + l) * HID + k];
        wo[k] = W_hh[(size_t)(3 * HID + l) * HID + k];
    }

    float h = 0.0f, c = 0.0f;

    // Software pipeline: x for step t is loaded during step t-1's compute.
    float x0 = xg[l +  0], x1 = xg[l + 32], x2 = xg[l + 64], x3 = xg[l + 96];

    for (int t = 0; t < T_STEPS; ++t) {
        float ai = x0, af = x1, ag = x2, ao = x3;
        if (t + 1 < T_STEPS) {
            const float* p = xg + (size_t)(t + 1) * GATES + l;
            x0 = p[0]; x1 = p[32]; x2 = p[64]; x3 = p[96];
        }

        // g = xg + h @ W_hh^T : broadcast h[k] (v_readlane -> SGPR), fmac.
#pragma unroll
        for (int k = 0; k < HID; ++k) {
            const float hk = readlane_f(h, k);
            ai = fmaf(hk, wi[k], ai);
            af = fmaf(hk, wf[k], af);
            ag = fmaf(hk, wg[k], ag);
            ao = fmaf(hk, wo[k], ao);
        }

        const float ig = sigmoid_fast(ai);
        const float fg = sigmoid_fast(af);
        const float gv = tanhf(ag);
        const float og = sigmoid_fast(ao);
        c = fg * c + ig * gv;
        h = og * tanhf(c);
    }

    // out = sigmoid(h . fc_w + fc_b) : 32-wide dot via shfl_xor reduction.
    float v = h * fc_w[l];
#pragma unroll
    for (int off = 16; off > 0; off >>= 1)
        v += __shfl_xor(v, off, 32);
    if (l == 0)
        out[0] = sigmoid_fast(v + fc_b[0]);
}

// ---------------------------------------------------------------------------
extern "C" void kernel_launch(void* const* d_in, const int* in_sizes, int n_in,
                              void* d_out, int out_size, void* d_ws, size_t ws_size,
                              hipStream_t stream) {
    const float* input = (const float*)d_in[0];  // [T, 86]
    const float* W_ih  = (const float*)d_in[1];  // [128, 86]
    const float* W_hh  = (const float*)d_in[2];  // [128, 32]
    const float* b_ih  = (const float*)d_in[3];  // [128]
    const float* b_hh  = (const float*)d_in[4];  // [128]
    const float* fc_w  = (const float*)d_in[5];  // [1, 32]
    const float* fc_b  = (const float*)d_in[6];  // [1]
    float* out = (float*)d_out;

    // Workspace: x_gates [T,128] fp32 = 128 MiB (fits in 192 MB L2).
    float* xg = (float*)d_ws;

    xgates_wmma_kernel<<<T_STEPS / 16, 256, 0, stream>>>(input, W_ih, b_ih, b_hh, xg);
    lstm_scan_kernel<<<1, 32, 0, stream>>>(xg, W_hh, fc_w, fc_b, out);
}